// GroupFormerDecoder_70841190580638
// MI455X (gfx1250) — compile-verified
//
#include <hip/hip_runtime.h>

#ifndef __has_builtin
#define __has_builtin(x) 0
#endif

// ---- problem constants (match reference) ----
#define BATCH 4
#define NB    9
#define HF    512
#define WF    512
#define VHW   256          // VH == VW
#define CCH   64
#define NSP   4096
#define NPIX  (VHW * VHW)  // 65536
#define EPSV  1e-16f

// Typed pointers for the CDNA5 async global->LDS builtin (params are int4*).
typedef int v4i_ __attribute__((ext_vector_type(4)));
typedef int v8i_ __attribute__((ext_vector_type(8)));
typedef __attribute__((address_space(1))) v4i_ as1_v4i;
typedef __attribute__((address_space(3))) v4i_ as3_v4i;
typedef __attribute__((address_space(3))) void as3_void;

// LDS byte-offset of a __shared__ object (generic -> AS3 -> int strips aperture).
static __device__ __forceinline__ unsigned lds_off_of(void* p) {
    return (unsigned)(unsigned long long)(as3_void*)p;
}

// ---------------------------------------------------------------------
// Tensor Data Mover: load a 2D tile (tile0 x tile1 elements of 4B) whose
// rows are stride0 elements apart in global memory, into contiguous LDS.
// D# packing per CDNA5 ISA (2-group form).
// ---------------------------------------------------------------------
static __device__ __forceinline__ void tdm_load_2d_f32(unsigned lds_off,
                                                       const void* gp,
                                                       unsigned tile0,
                                                       unsigned tile1,
                                                       unsigned long long stride0)
{
    unsigned long long ga = (unsigned long long)(size_t)gp;
    const unsigned td0 = 0x40000000u;   // tensor_dim0 (OOB bound; tile is in-bounds)
    const unsigned td1 = 16u;           // tensor_dim1

    v4i_ g0;
    g0[0] = 1;                                            // count=1, user descriptor
    g0[1] = (int)lds_off;                                 // lds_addr [63:32]
    g0[2] = (int)(unsigned)(ga & 0xffffffffu);            // global_addr[31:0]
    g0[3] = (int)((unsigned)((ga >> 32) & 0x01ffffffu)    // global_addr[56:32]
                  | 0x80000000u);                         // type=2 ("image") [127:126]

    v8i_ g1;
    g1[0] = (int)(2u << 16);                              // data_size=2 -> 4 bytes
    g1[1] = (int)((td0 & 0xffffu) << 16);                 // tensor_dim0[15:0]
    g1[2] = (int)((td0 >> 16) | ((td1 & 0xffffu) << 16)); // td0 hi | td1 lo
    g1[3] = (int)((td1 >> 16) | (tile0 << 16));           // td1 hi | tile_dim0
    g1[4] = (int)(tile1 & 0xffffu);                       // tile_dim1 (tile_dim2=0)
    g1[5] = (int)(unsigned)(stride0 & 0xffffffffu);       // tensor_dim0_stride lo
    g1[6] = (int)(unsigned)((stride0 >> 32) & 0xffffu);   // stride hi (dim1_stride=0)
    g1[7] = 0;

    asm volatile("tensor_load_to_lds %0, %1" :: "s"(g0), "s"(g1) : "memory");
}

static __device__ __forceinline__ void wait_tensorcnt0() {
#if __has_builtin(__builtin_amdgcn_s_wait_tensorcnt)
    __builtin_amdgcn_s_wait_tensorcnt(0);
#else
    asm volatile("s_wait_tensorcnt 0x0" ::: "memory");
#endif
}

// =====================================================================
// Kernel 1: antialiased 2x downsample (4-tap [1,3,3,1]/8 per dim, border
// renormalized like jax.image.resize antialias=True), exp, 9-way softmax,
// store attn_c, scatter-add into per-superpixel sums.
// One thread per (b, pixel).
// =====================================================================
__global__ void __launch_bounds__(256)
k_attn_sums(const float* __restrict__ aff,   // (B, 9, 512, 512)
            const int*   __restrict__ spix,  // (B, 9, 256, 256)
            float*       __restrict__ attn,  // ws: (B, 9, NPIX)
            float*       __restrict__ sums)  // ws: (B, NSP), pre-zeroed
{
    const int t = blockIdx.x * 256 + threadIdx.x;   // b * NPIX + p
    const int b = t >> 16;
    const int p = t & (NPIX - 1);
    const int y = p >> 8;
    const int x = p & 255;

    const float tw[4] = {1.f, 3.f, 3.f, 1.f};
    int   jy[4], jx[4];
    float wy[4], wx[4];
    float sy = 0.f, sx = 0.f;
#pragma unroll
    for (int tI = 0; tI < 4; ++tI) {
        int jyv = 2 * y - 1 + tI;
        int jxv = 2 * x - 1 + tI;
        bool vy = (jyv >= 0) && (jyv < HF);
        bool vx = (jxv >= 0) && (jxv < WF);
        jy[tI] = vy ? jyv : 0;
        jx[tI] = vx ? jxv : 0;
        wy[tI] = vy ? tw[tI] : 0.f;
        wx[tI] = vx ? tw[tI] : 0.f;
        sy += wy[tI];
        sx += wx[tI];
    }
    const float norm = 1.0f / (sy * sx);

    const float* ab = aff + (size_t)b * NB * HF * WF;

    float e[NB];
    float denom = 0.f;
#pragma unroll
    for (int k = 0; k < NB; ++k) {
        const float* plane = ab + (size_t)k * HF * WF;
        float m = 0.f;
#pragma unroll
        for (int ty = 0; ty < 4; ++ty) {
            const float* row = plane + (size_t)jy[ty] * WF;
            float r = 0.f;
#pragma unroll
            for (int tx = 0; tx < 4; ++tx)
                r = fmaf(wx[tx], row[jx[tx]], r);
            m = fmaf(wy[ty], r, m);
        }
        float ex = expf(m * norm);
        e[k] = ex;
        denom += ex;
    }

    const float inv = 1.0f / (denom + EPSV);  // attn_c = e / (denom + eps)
#pragma unroll
    for (int k = 0; k < NB; ++k) {
        float w = e[k] * inv;
        size_t off = ((size_t)b * NB + k) * NPIX + p;
        attn[off] = w;
        int s = spix[off];
        if ((unsigned)s < NSP)               // spix == NSP feeds only the dropped bin
            atomicAdd(&sums[b * NSP + s], w);
    }
}

// =====================================================================
// Kernel 2: per-edge renormalization + weighted scatter into out.
// One wave per pixel (8 waves / block).
//  - v rows (contiguous 2 KB) staged via async global->LDS b128 loads.
//  - attn/spix 9x8 tiles (row stride 256 KB!) staged via TDM 2D descriptors.
// Lane L owns channels L and L+32; 9 edge weights broadcast via shuffles.
// =====================================================================
__global__ void __launch_bounds__(256)
k_aggregate(const float* __restrict__ v,     // (B, NPIX, 64)
            const int*   __restrict__ spix,  // (B, 9, NPIX)
            const float* __restrict__ attn,  // (B, 9, NPIX)
            const float* __restrict__ sums,  // (B, NSP)
            float*       __restrict__ out)   // (B, NSP, 64), pre-zeroed
{
    __shared__ float lv[8 * CCH];            // 8 pixels x 64 channels (2 KB)
    __shared__ float la[NB * 8];             // attn tile: 9 rows x 8 pixels
    __shared__ int   ls[NB * 8];             // spix tile: 9 rows x 8 pixels

    const int tid  = threadIdx.x;
    const int wave = tid >> 5;
    const int lane = tid & 31;
    const int gpb  = blockIdx.x * 8;         // first (b*NPIX+p) slot of this block
    const int b    = gpb >> 16;
    const int p0   = gpb & (NPIX - 1);

    // --- async stage of 8 contiguous v rows: 512 floats = 128 x b128 ---
    const float* gsrc = v + (size_t)gpb * CCH;
    if (tid < 128) {
#if __has_builtin(__builtin_amdgcn_global_load_async_to_lds_b128)
        __builtin_amdgcn_global_load_async_to_lds_b128(
            (as1_v4i*)(gsrc + tid * 4),
            (as3_v4i*)(&lv[tid * 4]),
            0, 0);
#else
        ((float4*)lv)[tid] = ((const float4*)gsrc)[tid];
#endif
    }

    // --- TDM stage of the 9x8 attn / spix tiles (wave 0 issues, all wait) ---
    if (wave == 0) {
        const size_t ebase = (size_t)b * NB * NPIX + p0;
        tdm_load_2d_f32(lds_off_of(la), attn + ebase, 8u, NB, (unsigned long long)NPIX);
        tdm_load_2d_f32(lds_off_of(ls), spix + ebase, 8u, NB, (unsigned long long)NPIX);
        wait_tensorcnt0();
    }
#if __has_builtin(__builtin_amdgcn_s_wait_asynccnt)
    __builtin_amdgcn_s_wait_asynccnt(0);
#else
    asm volatile("s_wait_asynccnt 0" ::: "memory");
#endif
    __syncthreads();

    // --- lanes 0..8: per-edge weight w = attn_c / (sums_p[spix] + eps) ---
    float w = 0.f;
    int   s = -1;
    if (lane < NB) {
        float a  = la[lane * 8 + wave];
        int   si = ls[lane * 8 + wave];
        if ((unsigned)si < NSP) {
            s = si;
            w = a / (sums[b * NSP + si] + EPSV);
        }
    }

    const float v0 = lv[wave * CCH + lane];
    const float v1 = lv[wave * CCH + 32 + lane];
    float* ob = out + (size_t)b * NSP * CCH;

#pragma unroll
    for (int k = 0; k < NB; ++k) {
        float wk = __shfl(w, k);
        int   sk = __shfl(s, k);
        if (sk >= 0) {                        // wave-uniform branch
            float* o = ob + (size_t)sk * CCH;
            atomicAdd(o + lane,      wk * v0);  // coalesced: 32 lanes -> 128B RMW
            atomicAdd(o + lane + 32, wk * v1);
        }
    }
}

// =====================================================================
// Launch
// =====================================================================
extern "C" void kernel_launch(void* const* d_in, const int* in_sizes, int n_in,
                              void* d_out, int out_size, void* d_ws, size_t ws_size,
                              hipStream_t stream) {
    const float* aff  = (const float*)d_in[0];   // (4,9,512,512) f32
    const float* v    = (const float*)d_in[1];   // (4,256,256,64) f32
    const int*   spix = (const int*)d_in[2];     // (4,9,256,256) i32

    float* out  = (float*)d_out;                 // (4,4096,64) f32
    float* sums = (float*)d_ws;                  // B*NSP floats
    float* attn = sums + BATCH * NSP;            // B*9*NPIX floats (~9.4 MB)

    // Deterministic accumulation: zero accumulators every call (capture-safe).
    (void)hipMemsetAsync(out,  0, sizeof(float) * (size_t)BATCH * NSP * CCH, stream);
    (void)hipMemsetAsync(sums, 0, sizeof(float) * (size_t)BATCH * NSP, stream);

    k_attn_sums<<<(BATCH * NPIX) / 256, 256, 0, stream>>>(aff, spix, attn, sums);
    k_aggregate<<<(BATCH * NPIX) / 8, 256, 0, stream>>>(v, spix, attn, sums, out);

    (void)in_sizes; (void)n_in; (void)out_size; (void)ws_size;
}